// EdgeImportanceEncoder_12575664243129
// MI455X (gfx1250) — compile-verified
//
#include <hip/hip_runtime.h>
#include <hip/hip_bf16.h>

// Problem constants (fixed by the reference)
#define HEADS 8
#define NDIM  2048

typedef float v2f __attribute__((ext_vector_type(2)));
typedef float v8f __attribute__((ext_vector_type(8)));

// Fast sigmoid: v_exp_f32 + v_rcp_f32 (avoids IEEE div_scale/div_fixup expansion)
__device__ __forceinline__ float fast_sigmoid(float x) {
    return __builtin_amdgcn_rcpf(1.0f + __expf(-x));
}

// ---------------------------------------------------------------------------
// Kernel 1: zero the 128 MiB output with 128-bit stores (global_store_b128).
// ---------------------------------------------------------------------------
__global__ __launch_bounds__(256) void zero_out_kernel(float* __restrict__ out, int n) {
    int i4 = (blockIdx.x * blockDim.x + threadIdx.x) * 4;
    if (i4 + 3 < n) {
        *reinterpret_cast<float4*>(out + i4) = make_float4(0.f, 0.f, 0.f, 0.f);
    } else {
        for (int k = i4; k < n; ++k) out[k] = 0.f;
    }
}

// ---------------------------------------------------------------------------
// Kernel 2: per wave, process 16 edges:
//   D[16x16] = A[16x4](feats) * B[4x16](W padded) + C(bias b[h])  via
//   v_wmma_f32_16x16x4_f32, then sigmoid and scatter with
//   global_atomic_add_f32 into out[h, row, col].
// Scatter is redistributed across all 32 lanes: lane (n, half) handles
//   head h = n&7, edges  half*8 + (n>>3)*4 + {0..3}
// so in the full-group case every lane does 4 unpredicated atomics.
// ---------------------------------------------------------------------------
__global__ __launch_bounds__(256) void edge_bias_scatter_kernel(
        const int*   __restrict__ row,
        const int*   __restrict__ col,
        const float* __restrict__ score,
        const float* __restrict__ W,     // [3, HEADS] row-major
        const float* __restrict__ bvec,  // [HEADS]
        float*       __restrict__ out,   // [HEADS, NDIM, NDIM]
        int E) {
    const int lane  = threadIdx.x & 31;
    const int wave  = threadIdx.x >> 5;
    const int group = blockIdx.x * (blockDim.x >> 5) + wave;
    const int num_groups = (E + 15) >> 4;
    if (group >= num_groups) return;           // wave-uniform: EXEC stays all-1s

    const int base  = group << 4;
    const int count = min(16, E - base);

    const int n    = lane & 15;  // A: edge row m; B/D: head column n
    const int half = lane >> 4;  // K-half selector for A/B, M-half for D

    // ---- B matrix 4x16: lane n holds W[:, n], K split across lane halves ----
    v2f bm = {0.f, 0.f};
    if (n < HEADS) {
        if (half == 0) { bm.x = W[0 * HEADS + n]; bm.y = W[1 * HEADS + n]; }
        else           { bm.x = W[2 * HEADS + n]; bm.y = 0.f; }   // K=3 pad
    }

    // ---- A matrix 16x4: lane m holds feats[base+m], K split across halves ----
    int e = base + n;
    if (e >= E) e = E - 1;                     // clamp; tail rows masked below
    const int r = row[e];
    const int c = col[e];
    const int flat = r * NDIM + c;             // valid in lanes 0..15 (m = n)
    v2f am;
    if (half == 0) { am.x = (float)r;  am.y = (float)c; }   // K=0,1
    else           { am.x = score[e];  am.y = 0.f;       }  // K=2, K=3 pad

    // ---- C preloaded with per-head bias so WMMA fuses the "+ b" ----
    const float bb = (n < HEADS) ? bvec[n] : 0.f;
    v8f acc = {bb, bb, bb, bb, bb, bb, bb, bb};

    // D = A*B + C   (v_wmma_f32_16x16x4_f32)
    acc = __builtin_amdgcn_wmma_f32_16x16x4_f32(
        /*neg_a=*/false, am, /*neg_b=*/false, bm,
        /*c_mod=*/(short)0, acc, /*reuse_a=*/false, /*reuse_b=*/false);

    // ---- Sigmoid all 8 outputs unconditionally (full EXEC, no branches) ----
    float sg[8];
    #pragma unroll
    for (int v = 0; v < 8; ++v) sg[v] = fast_sigmoid(acc[v]);

    if (count == 16) {
        // ---- Fast path (all groups when E % 16 == 0): all 32 lanes active ----
        // Lane handles head h = n&7; q=1 lanes take edges 4..7 of lane L-8.
        const int q = n >> 3;                       // quarter selector
        const int h = n & 7;
        const int eBase = (half << 3) + (q << 2);   // first edge this lane owns
        float* outh = out + (size_t)h * (size_t)(NDIM * NDIM);
        #pragma unroll
        for (int j = 0; j < 4; ++j) {
            const int   f  = __shfl(flat, eBase + j, 32);   // edge flat index
            const float hi = __shfl(sg[4 + j], lane - 8, 32); // lane L-8's sg[4+j]
            const float val = q ? hi : sg[j];
            unsafeAtomicAdd(outh + f, val);          // global_atomic_add_f32
        }
    } else {
        // ---- Ragged tail: original masked layout (lanes n<8 active) ----
        const int mBase   = half << 3;
        const bool headOK = (n < HEADS);
        float* outh = out + (size_t)n * (size_t)(NDIM * NDIM);
        #pragma unroll
        for (int v = 0; v < 8; ++v) {
            const int mm = mBase + v;
            const int f  = __shfl(flat, mm, 32);
            if (headOK && mm < count) {
                unsafeAtomicAdd(outh + f, sg[v]);
            }
        }
    }
}

// ---------------------------------------------------------------------------
// Launch: zero-fill output, then one wave per 16-edge group.
// Inputs (setup_inputs order): row[E], col[E], score[E], W[3*H], b[H], H, N
// ---------------------------------------------------------------------------
extern "C" void kernel_launch(void* const* d_in, const int* in_sizes, int n_in,
                              void* d_out, int out_size, void* d_ws, size_t ws_size,
                              hipStream_t stream) {
    const int*   row   = (const int*)  d_in[0];
    const int*   col   = (const int*)  d_in[1];
    const float* score = (const float*)d_in[2];
    const float* W     = (const float*)d_in[3];
    const float* bvec  = (const float*)d_in[4];
    float* out = (float*)d_out;
    const int E = in_sizes[0];

    // 1) zero the [HEADS, NDIM, NDIM] accumulator
    const int zthreads = (out_size + 3) / 4;
    zero_out_kernel<<<(zthreads + 255) / 256, 256, 0, stream>>>(out, out_size);

    // 2) edge scatter: 8 waves (16-edge groups) per 256-thread block
    const int num_groups      = (E + 15) / 16;
    const int waves_per_block = 256 / 32;
    const int blocks          = (num_groups + waves_per_block - 1) / waves_per_block;
    edge_bias_scatter_kernel<<<blocks, 256, 0, stream>>>(row, col, score, W, bvec, out, E);
}